// PointNetPlusPlusEncoder_22900765622371
// MI455X (gfx1250) — compile-verified
//
#include <hip/hip_runtime.h>
#include <hip/hip_bf16.h>
#include <math.h>

typedef __attribute__((ext_vector_type(2))) float v2f;
typedef __attribute__((ext_vector_type(8))) float v8f;
typedef unsigned long long u64;
typedef unsigned int u32;

#define EPS_BN 1e-5f

// ---------- helpers ----------
__device__ __forceinline__ u64 pack_max_key(float d, u32 i) {
  // max by distance, ties -> smallest index (jnp.argmax picks first)
  return ((u64)__float_as_uint(d) << 32) | (u64)(0xFFFFFFFFu - i);
}
__device__ __forceinline__ u64 pack_min_key(float d, u32 i) {
  // min by distance, ties -> smallest index (top_k order)
  return ((u64)__float_as_uint(d) << 32) | (u64)i;
}

// ---------- param prep: pad weights, fold BN into scale/bias ----------
__global__ void prep_layer_kernel(const float* __restrict__ W, const float* __restrict__ b,
                                  const float* __restrict__ g, const float* __restrict__ be,
                                  const float* __restrict__ m, const float* __restrict__ v,
                                  float* __restrict__ wpad, float* __restrict__ es,
                                  float* __restrict__ eb, int cout, int cin, int cinp) {
  int t = blockIdx.x * blockDim.x + threadIdx.x;
  int total = cout * cinp;
  if (t < total) {
    int r = t / cinp, c = t % cinp;
    wpad[t] = (c < cin) ? W[r * cin + c] : 0.f;
  }
  if (t < cout) {
    float s = g[t] / sqrtf(v[t] + EPS_BN);
    es[t] = s;
    eb[t] = (b[t] - m[t]) * s + be[t];
  }
}

__global__ void zero_kernel(float* __restrict__ p, int n) {
  int t = blockIdx.x * blockDim.x + threadIdx.x;
  if (t < n) p[t] = 0.f;
}

// ---------- farthest point sampling ----------
// One block per batch. Per-thread register cache of coords + running min-dist.
template <int BLOCK, int PPT>
__global__ void fps_kernel(const float* __restrict__ pts, int stride, int N, int npoint,
                           float* __restrict__ newxyz) {
  int b = blockIdx.x, tid = threadIdx.x;
  const float* base = pts + (size_t)b * N * stride;
  float px[PPT], py[PPT], pz[PPT], dd[PPT];
  for (int ii = 0; ii < PPT; ++ii) {
    int i = tid + ii * BLOCK;
    if (i < N) {
      const float* p = base + (size_t)i * stride;
      px[ii] = p[0]; py[ii] = p[1]; pz[ii] = p[2];
    } else { px[ii] = py[ii] = pz[ii] = 0.f; }
    dd[ii] = 1e10f;
  }
  __shared__ u64 rk[BLOCK];
  __shared__ float cc[3];
  __shared__ int sfar;
  if (tid == 0) sfar = 0;   // deterministic start at index 0
  __syncthreads();
  for (int it = 0; it < npoint; ++it) {
    if (tid == 0) {
      const float* c = base + (size_t)sfar * stride;
      float cx = c[0], cy = c[1], cz = c[2];
      cc[0] = cx; cc[1] = cy; cc[2] = cz;
      float* o = newxyz + ((size_t)b * npoint + it) * 3;
      o[0] = cx; o[1] = cy; o[2] = cz;
    }
    __syncthreads();
    float cx = cc[0], cy = cc[1], cz = cc[2];
    u64 best = 0;
    for (int ii = 0; ii < PPT; ++ii) {
      int i = tid + ii * BLOCK;
      float dx = px[ii] - cx, dy = py[ii] - cy, dz = pz[ii] - cz;
      float d = dx * dx + dy * dy + dz * dz;
      float nd = fminf(dd[ii], d);
      dd[ii] = nd;
      if (i < N) {
        u64 k = pack_max_key(nd, (u32)i);
        if (k > best) best = k;
      }
    }
    rk[tid] = best;
    __syncthreads();
    for (int s = BLOCK / 2; s > 0; s >>= 1) {
      if (tid < s) { u64 o = rk[tid + s]; if (o > rk[tid]) rk[tid] = o; }
      __syncthreads();
    }
    if (tid == 0) sfar = (int)(0xFFFFFFFFu - (u32)(rk[0] & 0xFFFFFFFFu));
    __syncthreads();
  }
}

// ---------- kNN (k smallest squared distances; set semantics, order-free) ----------
// One block per (batch, query). Distances cached in registers; K selection passes.
template <int BLOCK, int PPT>
__global__ void knn_kernel(const float* __restrict__ pts, int stride, int N,
                           const float* __restrict__ cent, int Q, int K,
                           int* __restrict__ idxout) {
  int q = blockIdx.x % Q, b = blockIdx.x / Q, tid = threadIdx.x;
  const float* base = pts + (size_t)b * N * stride;
  const float* c = cent + ((size_t)b * Q + q) * 3;
  float cx = c[0], cy = c[1], cz = c[2];
  float d2r[PPT];
  for (int ii = 0; ii < PPT; ++ii) {
    int i = tid + ii * BLOCK;
    if (i < N) {
      const float* p = base + (size_t)i * stride;
      float dx = p[0] - cx, dy = p[1] - cy, dz = p[2] - cz;
      d2r[ii] = dx * dx + dy * dy + dz * dz;
    } else d2r[ii] = 1e30f;
  }
  constexpr int NW = BLOCK / 32;
  __shared__ u64 wk[NW];
  __shared__ u64 pick;
  int lane = tid & 31, wave = tid >> 5;
  u64 thresh = 0;
  for (int pass = 0; pass < K; ++pass) {
    u64 mymin = ~0ull;
    for (int ii = 0; ii < PPT; ++ii) {
      int i = tid + ii * BLOCK;
      if (i < N) {
        u64 k = pack_min_key(d2r[ii], (u32)i);
        if (k >= thresh && k < mymin) mymin = k;
      }
    }
    for (int off = 16; off > 0; off >>= 1) {
      u64 o = __shfl_xor(mymin, off, 32);
      if (o < mymin) mymin = o;
    }
    if (lane == 0) wk[wave] = mymin;
    __syncthreads();
    if (tid == 0) {
      u64 mm = wk[0];
      for (int w = 1; w < NW; ++w) if (wk[w] < mm) mm = wk[w];
      pick = mm;
      idxout[((size_t)b * Q + q) * K + pass] = (int)(u32)(mm & 0xFFFFFFFFu);
    }
    __syncthreads();
    thresh = pick + 1;  // keys unique -> strictly-greater exclusion
  }
}

// ---------- gather+group SA1: out (B, 8, 4096), ch0-2 rel xyz, ch3-5 feats, pad ----------
__global__ void group1_kernel(const float* __restrict__ pts, const float* __restrict__ nx,
                              const int* __restrict__ idx, float* __restrict__ g) {
  int t = blockIdx.x * blockDim.x + threadIdx.x;
  if (t >= 32 * 128 * 32) return;
  int b = t >> 12;
  int rem = t & 4095;        // pixel = q*32+s
  int q = rem >> 5;
  int i = idx[t];
  const float* p = pts + ((size_t)b * 16384 + i) * 6;
  const float* c = nx + ((size_t)(b * 128 + q)) * 3;
  float* gb = g + (size_t)b * 8 * 4096;
  gb[0 * 4096 + rem] = p[0] - c[0];
  gb[1 * 4096 + rem] = p[1] - c[1];
  gb[2 * 4096 + rem] = p[2] - c[2];
  gb[3 * 4096 + rem] = p[3];
  gb[4 * 4096 + rem] = p[4];
  gb[5 * 4096 + rem] = p[5];
  gb[6 * 4096 + rem] = 0.f;
  gb[7 * 4096 + rem] = 0.f;
}

// ---------- gather+group SA2: out (B, 132, 2048); faithful channel-axis gather quirk ----------
__global__ void group2_kernel(const float* __restrict__ xyz1, const float* __restrict__ nx2,
                              const float* __restrict__ feat1, const int* __restrict__ idx,
                              float* __restrict__ g) {
  int bid = blockIdx.x;      // = (b*32+q)*64 + j
  int j = bid & 63;
  int bq = bid >> 6;
  int q = bq & 31;
  int b = bq >> 5;
  int tid = threadIdx.x;     // 128 threads
  int i = idx[bid];
  int pix = q * 64 + j;
  float* gb = g + (size_t)b * 132 * 2048;
  // quirk: features (B, C=128, P=128) indexed along dim-1 by point index i
  gb[(size_t)(3 + tid) * 2048 + pix] = feat1[((size_t)b * 128 + i) * 128 + tid];
  if (tid < 3) {
    gb[(size_t)tid * 2048 + pix] =
        xyz1[((size_t)b * 128 + i) * 3 + tid] - nx2[((size_t)b * 32 + q) * 3 + tid];
  } else if (tid == 3) {
    gb[(size_t)131 * 2048 + pix] = 0.f;  // pad channel
  }
}

// ---------- fused 3-layer MLP (WMMA f32 16x16x4) + BN + ReLU + sample max-pool ----------
// X: (B, CIN0P, PIX) channel-major. Each wave owns one 16-pixel tile (within one group).
// D=W*X: M=cout tile (A = weights 16x4), N=16 pixels (B operand), K=cin chunks of 4.
template <int CIN0P, int C1, int C2, int C3, int SGRP, int WAVES>
__global__ void mlp_kernel(const float* __restrict__ X, const float* __restrict__ Wp,
                           const float* __restrict__ eff, float* __restrict__ outF, int PIX) {
  constexpr int HMAX = (C1 > C2) ? C1 : C2;
  constexpr int T1 = C1 / 16, T2 = C2 / 16, T3 = C3 / 16;
  __shared__ float hbuf[WAVES][HMAX][16];  // wave-private activation staging
  int b = blockIdx.y;
  int wave = (int)threadIdx.x >> 5;
  int lane = (int)threadIdx.x & 31;
  int lh = lane >> 4;        // lane half: K-split / M-split selector
  int n = lane & 15;         // N (pixel) index, also A-row index
  int pix0 = (blockIdx.x * WAVES + wave) * 16;
  const float* Xb = X + (size_t)b * CIN0P * PIX;
  const float* W1 = Wp;
  const float* W2 = Wp + C1 * CIN0P;
  const float* W3 = Wp + C1 * CIN0P + C2 * C1;
  const float* s1 = eff;                     const float* b1 = eff + C1;
  const float* s2 = eff + 2 * C1;            const float* b2 = eff + 2 * C1 + C2;
  const float* s3 = eff + 2 * C1 + 2 * C2;   const float* b3 = eff + 2 * C1 + 2 * C2 + C3;

  v8f zero;
  for (int e = 0; e < 8; ++e) zero[e] = 0.f;
  v8f acc[T3];

  // ---- layer 1: X from global ----
  for (int t = 0; t < T1; ++t) acc[t] = zero;
  for (int c = 0; c < CIN0P / 4; ++c) {
    int k0 = 4 * c + 2 * lh;
    v2f bv;
    bv.x = Xb[(size_t)k0 * PIX + pix0 + n];
    bv.y = Xb[(size_t)(k0 + 1) * PIX + pix0 + n];
    for (int t = 0; t < T1; ++t) {
      const float* wr = W1 + (size_t)(t * 16 + n) * CIN0P + k0;
      v2f av; av.x = wr[0]; av.y = wr[1];
      acc[t] = __builtin_amdgcn_wmma_f32_16x16x4_f32(false, av, false, bv,
                                                     (short)0, acc[t], false, false);
    }
  }
  for (int t = 0; t < T1; ++t)
    for (int j = 0; j < 8; ++j) {
      int m = t * 16 + 8 * lh + j;
      hbuf[wave][m][n] = fmaxf(acc[t][j] * s1[m] + b1[m], 0.f);
    }
  __syncthreads();

  // ---- layer 2: X from LDS ----
  for (int t = 0; t < T2; ++t) acc[t] = zero;
  for (int c = 0; c < C1 / 4; ++c) {
    int k0 = 4 * c + 2 * lh;
    v2f bv;
    bv.x = hbuf[wave][k0][n];
    bv.y = hbuf[wave][k0 + 1][n];
    for (int t = 0; t < T2; ++t) {
      const float* wr = W2 + (size_t)(t * 16 + n) * C1 + k0;
      v2f av; av.x = wr[0]; av.y = wr[1];
      acc[t] = __builtin_amdgcn_wmma_f32_16x16x4_f32(false, av, false, bv,
                                                     (short)0, acc[t], false, false);
    }
  }
  __syncthreads();
  for (int t = 0; t < T2; ++t)
    for (int j = 0; j < 8; ++j) {
      int m = t * 16 + 8 * lh + j;
      hbuf[wave][m][n] = fmaxf(acc[t][j] * s2[m] + b2[m], 0.f);
    }
  __syncthreads();

  // ---- layer 3 ----
  for (int t = 0; t < T3; ++t) acc[t] = zero;
  for (int c = 0; c < C2 / 4; ++c) {
    int k0 = 4 * c + 2 * lh;
    v2f bv;
    bv.x = hbuf[wave][k0][n];
    bv.y = hbuf[wave][k0 + 1][n];
    for (int t = 0; t < T3; ++t) {
      const float* wr = W3 + (size_t)(t * 16 + n) * C2 + k0;
      v2f av; av.x = wr[0]; av.y = wr[1];
      acc[t] = __builtin_amdgcn_wmma_f32_16x16x4_f32(false, av, false, bv,
                                                     (short)0, acc[t], false, false);
    }
  }

  // ---- BN+ReLU, max over the tile's 16 samples, atomic-max into (B,C3,PIX/SGRP) ----
  int p = pix0 / SGRP;          // group index (tile never straddles a group)
  int POUT = PIX / SGRP;
  for (int t = 0; t < T3; ++t)
    for (int j = 0; j < 8; ++j) {
      int m = t * 16 + 8 * lh + j;
      float v = fmaxf(acc[t][j] * s3[m] + b3[m], 0.f);
      for (int off = 1; off < 16; off <<= 1) v = fmaxf(v, __shfl_xor(v, off, 32));
      if (n == 0) {
        // ReLU output >= 0 -> int compare == float compare; buffer zero-initialized
        atomicMax((int*)(outF + ((size_t)b * C3 + m) * POUT + p), __float_as_int(v));
      }
    }
}

// ---------- final: out[b][p] = max over 256 channels (faithful to reference transpose) ----------
__global__ void final_max_kernel(const float* __restrict__ feat2, float* __restrict__ out) {
  int t = blockIdx.x * blockDim.x + threadIdx.x;
  if (t >= 32 * 32) return;
  int b = t >> 5, p = t & 31;
  const float* f = feat2 + (size_t)b * 256 * 32 + p;
  float m = -INFINITY;
  for (int c = 0; c < 256; ++c) m = fmaxf(m, f[(size_t)c * 32]);
  out[t] = m;
}

// ---------- host-side orchestration ----------
extern "C" void kernel_launch(void* const* d_in, const int* in_sizes, int n_in,
                              void* d_out, int out_size, void* d_ws, size_t ws_size,
                              hipStream_t stream) {
  (void)in_sizes; (void)n_in; (void)out_size; (void)ws_size;
  const float* points = (const float*)d_in[0];
  float* ws = (float*)d_ws;

  size_t o = 0;
  float* wpad1 = ws + o; o += 64 * 8 + 64 * 64 + 128 * 64;
  float* wpad2 = ws + o; o += 128 * 132 + 128 * 128 + 256 * 128;
  float* eff1 = ws + o;  o += 2 * (64 + 64 + 128);
  float* eff2 = ws + o;  o += 2 * (128 + 128 + 256);
  float* nx1 = ws + o;   o += (size_t)32 * 128 * 3;
  float* nx2 = ws + o;   o += (size_t)32 * 32 * 3;
  int* idx1 = (int*)(ws + o); o += (size_t)32 * 128 * 32;
  int* idx2 = (int*)(ws + o); o += (size_t)32 * 32 * 64;
  float* g1 = ws + o;    o += (size_t)32 * 8 * 4096;
  float* g2 = ws + o;    o += (size_t)32 * 132 * 2048;
  float* f1 = ws + o;    o += (size_t)32 * 128 * 128;
  float* f2 = ws + o;    o += (size_t)32 * 256 * 32;

  // ---- param prep (d_in: 0=points, 1..18 = sa1 3x(W,b,g,be,m,v), 19..36 = sa2) ----
  {
    int cout1[3] = {64, 64, 128}, cin1[3] = {6, 64, 64}, cinp1[3] = {8, 64, 64};
    float* wp = wpad1; float* ep = eff1;
    for (int l = 0; l < 3; ++l) {
      int base = 1 + l * 6;
      int tot = cout1[l] * cinp1[l];
      prep_layer_kernel<<<(tot + 255) / 256, 256, 0, stream>>>(
          (const float*)d_in[base + 0], (const float*)d_in[base + 1],
          (const float*)d_in[base + 2], (const float*)d_in[base + 3],
          (const float*)d_in[base + 4], (const float*)d_in[base + 5],
          wp, ep, ep + cout1[l], cout1[l], cin1[l], cinp1[l]);
      wp += tot; ep += 2 * cout1[l];
    }
    int cout2[3] = {128, 128, 256}, cin2[3] = {131, 128, 128}, cinp2[3] = {132, 128, 128};
    wp = wpad2; ep = eff2;
    for (int l = 0; l < 3; ++l) {
      int base = 19 + l * 6;
      int tot = cout2[l] * cinp2[l];
      prep_layer_kernel<<<(tot + 255) / 256, 256, 0, stream>>>(
          (const float*)d_in[base + 0], (const float*)d_in[base + 1],
          (const float*)d_in[base + 2], (const float*)d_in[base + 3],
          (const float*)d_in[base + 4], (const float*)d_in[base + 5],
          wp, ep, ep + cout2[l], cout2[l], cin2[l], cinp2[l]);
      wp += tot; ep += 2 * cout2[l];
    }
  }

  zero_kernel<<<(32 * 128 * 128 + 255) / 256, 256, 0, stream>>>(f1, 32 * 128 * 128);
  zero_kernel<<<(32 * 256 * 32 + 255) / 256, 256, 0, stream>>>(f2, 32 * 256 * 32);

  // ---- SA1 ----
  fps_kernel<512, 32><<<32, 512, 0, stream>>>(points, 6, 16384, 128, nx1);
  knn_kernel<256, 64><<<32 * 128, 256, 0, stream>>>(points, 6, 16384, nx1, 128, 32, idx1);
  group1_kernel<<<(32 * 128 * 32 + 255) / 256, 256, 0, stream>>>(points, nx1, idx1, g1);
  mlp_kernel<8, 64, 64, 128, 32, 8><<<dim3(4096 / (16 * 8), 32), 256, 0, stream>>>(
      g1, wpad1, eff1, f1, 4096);

  // ---- SA2 ----
  fps_kernel<128, 1><<<32, 128, 0, stream>>>(nx1, 3, 128, 32, nx2);
  knn_kernel<128, 1><<<32 * 32, 128, 0, stream>>>(nx1, 3, 128, nx2, 32, 64, idx2);
  group2_kernel<<<32 * 32 * 64, 128, 0, stream>>>(nx1, nx2, f1, idx2, g2);
  mlp_kernel<132, 128, 128, 256, 64, 4><<<dim3(2048 / (16 * 4), 32), 128, 0, stream>>>(
      g2, wpad2, eff2, f2, 2048);

  // ---- final channel-max ----
  final_max_kernel<<<4, 256, 0, stream>>>(f2, (float*)d_out);
}